// OutputLayer_53798760349842
// MI455X (gfx1250) — compile-verified
//
#include <hip/hip_runtime.h>
#include <hip/hip_bf16.h>

// Problem sizes (match reference)
#define N_I 500000
#define N_H 200000
#define F_I 32
#define F_H 16
#define EF  8
#define E1  2000000
#define E2  2000000

typedef __attribute__((ext_vector_type(2))) float v2f;
typedef __attribute__((ext_vector_type(8))) float v8f;

// ---------------------------------------------------------------------------
// Kernel 0: assemble padded B matrices for the WMMA node GEMMs.
//  Bi [32 x 16]: cols 0..7 = W_edge_i2i[:,k], col 8 = b_edge_i2i,
//                col 9 = W_root_h2i + W_root_i2i, cols 10..15 = 0
//  Bh [16 x 16]: cols 0..7 = W_edge_h2i[:,k], col 8 = b_edge_h2i, rest 0
// Stored row-major as B[f*16 + n] (K-major, matching the WMMA B layout loads).
// ---------------------------------------------------------------------------
__global__ void build_B_kernel(const float* __restrict__ We_i, const float* __restrict__ be_i,
                               const float* __restrict__ Wr_h, const float* __restrict__ Wr_i,
                               const float* __restrict__ We_h, const float* __restrict__ be_h,
                               float* __restrict__ Bi, float* __restrict__ Bh) {
    int t = threadIdx.x;             // 512 threads
    if (t < F_I * 16) {              // Bi
        int f = t >> 4, n = t & 15;
        float v = 0.0f;
        if (n < 8)       v = We_i[f * 8 + n];
        else if (n == 8) v = be_i[f];
        else if (n == 9) v = Wr_h[f] + Wr_i[f];
        Bi[t] = v;
    }
    if (t < F_H * 16) {              // Bh
        int f = t >> 4, n = t & 15;
        float v = 0.0f;
        if (n < 8)       v = We_h[f * 8 + n];
        else if (n == 8) v = be_h[f];
        Bh[t] = v;
    }
}

// ---------------------------------------------------------------------------
// Kernels 1/2: node precompute GEMM  U[N,16] = x[N,F] @ B[F,16]
// One wave32 per tile of 16 nodes; f32 WMMA 16x16x4, K loop over F.
//
// A 16x4 f32 layout (2 VGPRs): lane L<16 holds A[M=L, K={k0,k0+1}],
//                              lane L>=16 holds A[M=L-16, K={k0+2,k0+3}].
// B 4x16 f32 layout (2 VGPRs): lane L<16 holds B[K={k0,k0+1}, N=L],
//                              lane L>=16 holds B[K={k0+2,k0+3}, N=L-16].
// D 16x16 f32 (8 VGPRs): vgpr r = row M=r (lanes 0-15) / M=r+8 (lanes 16-31),
//                        N = lane&15.
// ---------------------------------------------------------------------------
template <int F>
__global__ __launch_bounds__(256)
void node_precompute_kernel(const float* __restrict__ x, const float* __restrict__ Bmat,
                            float* __restrict__ U, int n_tiles) {
    int wave = (int)((blockIdx.x * blockDim.x + threadIdx.x) >> 5); // wave-uniform
    int lane = threadIdx.x & 31;
    if (wave >= n_tiles) return;                                   // whole-wave exit

    const int m    = lane & 15;
    const int half = lane >> 4;          // 0 or 1 -> K offset 0 or 2
    const int node0 = wave * 16;
    const float* xrow = x + (size_t)(node0 + m) * F;

    v8f acc = {0.f, 0.f, 0.f, 0.f, 0.f, 0.f, 0.f, 0.f};

#pragma unroll
    for (int k0 = 0; k0 < F; k0 += 4) {
        const int ka = k0 + half * 2;
        v2f a, b;
        a.x = xrow[ka];
        a.y = xrow[ka + 1];
        b.x = Bmat[ka * 16 + m];
        b.y = Bmat[(ka + 1) * 16 + m];
        acc = __builtin_amdgcn_wmma_f32_16x16x4_f32(
            /*neg_a=*/false, a, /*neg_b=*/false, b,
            /*c_mod=*/(short)0, acc, /*reuse_a=*/false, /*reuse_b=*/false);
    }

    float* Ubase = U + (size_t)node0 * 16;
#pragma unroll
    for (int r = 0; r < 8; ++r) {
        // element (M = r + half*8, N = m)
        Ubase[(size_t)(r + half * 8) * 16 + m] = acc[r];
    }
}

// ---------------------------------------------------------------------------
// Kernel 3: initialize output with root term (resets out every call).
// out[n] = U_i[n,9] + bias_h2i + bias_i2i
// ---------------------------------------------------------------------------
__global__ __launch_bounds__(256)
void init_out_kernel(const float* __restrict__ Ui, const float* __restrict__ bias_h,
                     const float* __restrict__ bias_i, float* __restrict__ out, int n) {
    int i = blockIdx.x * blockDim.x + threadIdx.x;
    if (i < n) out[i] = Ui[(size_t)i * 16 + 9] + bias_h[0] + bias_i[0];
}

// ---------------------------------------------------------------------------
// Kernels 4/5: per-edge message + scatter-add.
// msg = ea[e,0:8] . U[src,0:8] + U[src,8];  atomicAdd(out[dst], msg)
// ---------------------------------------------------------------------------
__global__ __launch_bounds__(256)
void edge_scatter_kernel(const float* __restrict__ ea, const int* __restrict__ src,
                         const int* __restrict__ dst, const float* __restrict__ U,
                         float* __restrict__ out, int n_edges) {
    int e = blockIdx.x * blockDim.x + threadIdx.x;
    if (e >= n_edges) return;

    const float* a = ea + (size_t)e * 8;
    float4 a0 = *(const float4*)(a);
    float4 a1 = *(const float4*)(a + 4);

    const float* u = U + (size_t)src[e] * 16;
    float4 u0 = *(const float4*)(u);
    float4 u1 = *(const float4*)(u + 4);
    float  c  = u[8];

    float m = c
            + a0.x * u0.x + a0.y * u0.y + a0.z * u0.z + a0.w * u0.w
            + a1.x * u1.x + a1.y * u1.y + a1.z * u1.z + a1.w * u1.w;

    atomicAdd(out + dst[e], m);
}

// ---------------------------------------------------------------------------
extern "C" void kernel_launch(void* const* d_in, const int* in_sizes, int n_in,
                              void* d_out, int out_size, void* d_ws, size_t ws_size,
                              hipStream_t stream) {
    const float* x_indivi   = (const float*)d_in[0];
    const float* x_house    = (const float*)d_in[1];
    const float* ea_h2i     = (const float*)d_in[2];
    const float* ea_i2i     = (const float*)d_in[3];
    const float* W_edge_h2i = (const float*)d_in[4];
    const float* b_edge_h2i = (const float*)d_in[5];
    const float* W_edge_i2i = (const float*)d_in[6];
    const float* b_edge_i2i = (const float*)d_in[7];
    const float* W_root_h2i = (const float*)d_in[8];
    const float* bias_h2i   = (const float*)d_in[9];
    const float* W_root_i2i = (const float*)d_in[10];
    const float* bias_i2i   = (const float*)d_in[11];
    const int*   src_h2i    = (const int*)d_in[12];
    const int*   dst_h2i    = (const int*)d_in[13];
    const int*   src_i2i    = (const int*)d_in[14];
    const int*   dst_i2i    = (const int*)d_in[15];
    float*       out        = (float*)d_out;

    // Workspace layout
    char*  ws = (char*)d_ws;
    float* Bi = (float*)ws;                                  // 32*16 floats
    float* Bh = (float*)(ws + 2048);                         // 16*16 floats
    float* Ui = (float*)(ws + 4096);                         // N_I*16 floats (32 MB)
    float* Uh = (float*)(ws + 4096 + (size_t)N_I * 16 * 4);  // N_H*16 floats (12.8 MB)

    // 0: build padded weight matrices
    build_B_kernel<<<1, 512, 0, stream>>>(W_edge_i2i, b_edge_i2i, W_root_h2i, W_root_i2i,
                                          W_edge_h2i, b_edge_h2i, Bi, Bh);

    // 1: U_i = x_indivi @ Bi   (31250 tiles of 16 nodes, 1 wave each, 8 waves/block)
    {
        int tiles = N_I / 16;                     // 31250
        int blocks = (tiles + 7) / 8;             // 256 threads = 8 wave32
        node_precompute_kernel<F_I><<<blocks, 256, 0, stream>>>(x_indivi, Bi, Ui, tiles);
    }
    // 2: U_h = x_house @ Bh
    {
        int tiles = N_H / 16;                     // 12500
        int blocks = (tiles + 7) / 8;
        node_precompute_kernel<F_H><<<blocks, 256, 0, stream>>>(x_house, Bh, Uh, tiles);
    }

    // 3: out = root term (also resets accumulator each call)
    init_out_kernel<<<(N_I + 255) / 256, 256, 0, stream>>>(Ui, bias_h2i, bias_i2i, out, N_I);

    // 4: h2i edges
    edge_scatter_kernel<<<(E1 + 255) / 256, 256, 0, stream>>>(ea_h2i, src_h2i, dst_h2i, Uh, out, E1);
    // 5: i2i edges
    edge_scatter_kernel<<<(E2 + 255) / 256, 256, 0, stream>>>(ea_i2i, src_i2i, dst_i2i, Ui, out, E2);
}